// FokkerPlanck2DBase_66048007078447
// MI455X (gfx1250) — compile-verified
//
#include <hip/hip_runtime.h>
#include <stdint.h>

// Problem constants (match reference: B=16, NX=NY=1024, fp32).
#define NXc 1024
#define NYc 1024
#define NB  16

// Tiling: 16 (x) x 64 (y) outputs per 256-thread block; halo of 1.
#define TX   16
#define TY   64
#define HX   (TX + 2)   // 18 rows
#define JH   4          // LDS col of gy==y0 (halo-left at col 3, halo-right at col 68)
#define LDSP 72         // padded LDS row stride (floats); keeps b128 LDS stores 16B-aligned
#define TPB  256
#define PLANE (HX * LDSP)  // 1296 floats per staged plane

#define OFF_A0 0
#define OFF_A1 (1 * PLANE)
#define OFF_B0 (2 * PLANE)
#define OFF_B1 (3 * PLANE)
#define OFF_B2 (4 * PLANE)
#define OFF_F0 (5 * PLANE)
#define OFF_F1 (6 * PLANE)
#define SMEM_FLOATS (7 * PLANE)   // ~36.3 KB of the 320 KB/WGP LDS

// ---------- CDNA5 async global->LDS helpers ----------

__device__ __forceinline__ uint32_t lds_byte_off(const void* p) {
  // generic -> LDS(addrspace 3) cast; ptrtoint gives the 32-bit LDS byte offset
  return (uint32_t)(uintptr_t)(__attribute__((address_space(3))) const void*)p;
}

__device__ __forceinline__ void async_ld_b32(const void* g, uint32_t lds_off) {
  asm volatile("global_load_async_to_lds_b32 %0, %1, off"
               :: "v"(lds_off), "v"((unsigned long long)(uintptr_t)g)
               : "memory");
}

__device__ __forceinline__ void async_ld_b128(const void* g, uint32_t lds_off) {
  // 16B per lane, global -> LDS, tracked on ASYNCcnt
  asm volatile("global_load_async_to_lds_b128 %0, %1, off"
               :: "v"(lds_off), "v"((unsigned long long)(uintptr_t)g)
               : "memory");
}

__device__ __forceinline__ void wait_async_0() {
#if __has_builtin(__builtin_amdgcn_s_wait_asynccnt)
  __builtin_amdgcn_s_wait_asynccnt(0);
#else
  asm volatile("s_wait_asynccnt 0" ::: "memory");
#endif
}

// ---------- slow path (domain-edge points, exact one-sided stencils) ----------

__device__ __forceinline__ float PV(const float* __restrict__ fb,
                                    const float* __restrict__ g, int x, int y) {
  return fb[x * NYc + y] * g[x * NYc + y];
}

__device__ __forceinline__ float d1x_g(const float* fb, const float* g, int x, int y) {
  if (x == 0)       return 0.5f * (-3.f * PV(fb,g,0,y) + 4.f * PV(fb,g,1,y) - PV(fb,g,2,y));
  if (x == NXc - 1) return 0.5f * ( 3.f * PV(fb,g,x,y) - 4.f * PV(fb,g,x-1,y) + PV(fb,g,x-2,y));
  return 0.5f * (PV(fb,g,x+1,y) - PV(fb,g,x-1,y));
}
__device__ __forceinline__ float d1y_g(const float* fb, const float* g, int x, int y) {
  if (y == 0)       return 0.5f * (-3.f * PV(fb,g,x,0) + 4.f * PV(fb,g,x,1) - PV(fb,g,x,2));
  if (y == NYc - 1) return 0.5f * ( 3.f * PV(fb,g,x,y) - 4.f * PV(fb,g,x,y-1) + PV(fb,g,x,y-2));
  return 0.5f * (PV(fb,g,x,y+1) - PV(fb,g,x,y-1));
}
__device__ __forceinline__ float d2x_g(const float* fb, const float* g, int x, int y) {
  if (x == 0)       return 2.f*PV(fb,g,0,y) - 5.f*PV(fb,g,1,y) + 4.f*PV(fb,g,2,y) - PV(fb,g,3,y);
  if (x == NXc - 1) return 2.f*PV(fb,g,x,y) - 5.f*PV(fb,g,x-1,y) + 4.f*PV(fb,g,x-2,y) - PV(fb,g,x-3,y);
  return PV(fb,g,x+1,y) - 2.f*PV(fb,g,x,y) + PV(fb,g,x-1,y);
}
__device__ __forceinline__ float d2y_g(const float* fb, const float* g, int x, int y) {
  if (y == 0)       return 2.f*PV(fb,g,x,0) - 5.f*PV(fb,g,x,1) + 4.f*PV(fb,g,x,2) - PV(fb,g,x,3);
  if (y == NYc - 1) return 2.f*PV(fb,g,x,y) - 5.f*PV(fb,g,x,y-1) + 4.f*PV(fb,g,x,y-2) - PV(fb,g,x,y-3);
  return PV(fb,g,x,y+1) - 2.f*PV(fb,g,x,y) + PV(fb,g,x,y-1);
}
// d/dx(d/dy P); 1D operators on distinct axes commute, so dxdy == dydx exactly.
__device__ __forceinline__ float dxdy_g(const float* fb, const float* g, int x, int y) {
  if (x == 0)       return 0.5f * (-3.f * d1y_g(fb,g,0,y) + 4.f * d1y_g(fb,g,1,y) - d1y_g(fb,g,2,y));
  if (x == NXc - 1) return 0.5f * ( 3.f * d1y_g(fb,g,x,y) - 4.f * d1y_g(fb,g,x-1,y) + d1y_g(fb,g,x-2,y));
  return 0.5f * (d1y_g(fb,g,x+1,y) - d1y_g(fb,g,x-1,y));
}

// Rare (~0.4% of points); keep out of the hot loop's code footprint.
__device__ __attribute__((noinline)) float slow_df(const float* fb,
                                                   const float* A0, const float* A1,
                                                   const float* B0, const float* B1,
                                                   const float* B2, int x, int y) {
  float gA = d1x_g(fb, A0, x, y) + d1y_g(fb, A1, x, y);
  float gB = d2x_g(fb, B0, x, y) + d2y_g(fb, B1, x, y) + 2.f * dxdy_g(fb, B2, x, y);
  return 0.5f * gB - gA;
}

// ---------- main kernel ----------

#define SM(o, i, jj) smem[(o) + (i) * LDSP + (jj)]

__global__ __launch_bounds__(TPB) void fp2d_step(const float* __restrict__ f,
                                                 const float* __restrict__ Ag,
                                                 const float* __restrict__ Bg,
                                                 const float* __restrict__ dt,
                                                 float* __restrict__ out) {
  __shared__ float smem[SMEM_FLOATS];
  const int tid = threadIdx.x;
  const int x0  = blockIdx.y * TX;
  const int y0  = blockIdx.x * TY;
  const uint32_t lbase = lds_byte_off(&smem[0]);

  const float* A0 = Ag;
  const float* A1 = Ag + NXc * NYc;
  const float* B0 = Bg;
  const float* B1 = Bg + (size_t)NXc * NYc;
  const float* B2 = Bg + 2 * (size_t)NXc * NYc;

  // ---- Precompute staging slots (batch-invariant tile geometry).
  // Interior 18x64 region: 288 b128 chunks (16B-aligned global AND LDS sides).
  //   slot0: chunk = tid (all threads), slot1: chunk = tid+256 (tid < 32).
  // Halo columns (gy = y0-1 and y0+64): 36 b32 elems -> threads 32..67.
  uint32_t g0, l0, g1 = 0, l1 = 0;
  {
    int li = tid >> 4, c = tid & 15;
    int gx = min(max(x0 - 1 + li, 0), NXc - 1);        // x-clamped rows feed slow-path pts only
    g0 = 4u * (uint32_t)(gx * NYc + y0 + 4 * c);       // interior cols: no y clamp needed
    l0 = 4u * (uint32_t)(li * LDSP + JH + 4 * c);      // 16B-aligned (LDSP=72, JH=4)
  }
  const bool s1_128 = (tid < 32);
  const bool s1_32  = (tid >= 32) && (tid < 32 + 36);
  if (s1_128) {
    int ch = tid + 256;                                 // chunks 256..287 -> rows 16..17
    int li = ch >> 4, c = ch & 15;
    int gx = min(max(x0 - 1 + li, 0), NXc - 1);
    g1 = 4u * (uint32_t)(gx * NYc + y0 + 4 * c);
    l1 = 4u * (uint32_t)(li * LDSP + JH + 4 * c);
  } else if (s1_32) {
    int e = tid - 32;
    int li = e >> 1, side = e & 1;
    int gx  = min(max(x0 - 1 + li, 0), NXc - 1);
    int gy  = min(max(side ? (y0 + TY) : (y0 - 1), 0), NYc - 1);
    int col = side ? (JH + TY) : (JH - 1);              // cols 68 / 3
    g1 = 4u * (uint32_t)(gx * NYc + gy);
    l1 = 4u * (uint32_t)(li * LDSP + col);
  }

  // One-tile async stage: <=2 instructions per thread.
  auto stage = [&](const void* base, uint32_t plane_bytes) {
    async_ld_b128((const char*)base + g0, lbase + plane_bytes + l0);
    if (s1_128)     async_ld_b128((const char*)base + g1, lbase + plane_bytes + l1);
    else if (s1_32) async_ld_b32 ((const char*)base + g1, lbase + plane_bytes + l1);
  };

  // ---- Stage batch-invariant A/B tiles once (reused for all 16 batches).
  {
    const float* planes[5] = {A0, A1, B0, B1, B2};
    #pragma unroll
    for (int c = 0; c < 5; ++c)
      stage(planes[c], 4u * (uint32_t)(c * PLANE));
  }
  // ---- Stage f tile for batch 0 into buffer 0.
  stage(f, 4u * (uint32_t)OFF_F0);
  wait_async_0();
  __syncthreads();

  // Thread -> points: one x row, 4 consecutive y per thread (float4 store).
  const int trow = tid >> 4;          // 0..15
  const int tcol = tid & 15;          // 0..15
  const int i    = trow + 1;          // local x with halo
  const int j0   = JH + 4 * tcol;     // local y (gy0 maps to col JH + 4*tcol)
  const int gx   = x0 + trow;
  const int gy0  = y0 + 4 * tcol;
  const bool xedge = (gx == 0) || (gx == NXc - 1);

  #pragma unroll 1
  for (int b = 0; b < NB; ++b) {
    const int cur  = b & 1;
    const int foff = cur ? OFF_F1 : OFF_F0;

    // Prefetch next batch's f tile into the other buffer (overlaps with compute).
    if (b + 1 < NB)
      stage(f + (size_t)(b + 1) * NXc * NYc,
            4u * (uint32_t)(cur ? OFF_F0 : OFF_F1));

    const float dtb = dt[b];
    const float* fb = f + (size_t)b * NXc * NYc;
    float* ob = out + (size_t)b * NXc * NYc;

    float res[4];
    #pragma unroll
    for (int q = 0; q < 4; ++q) {
      const int j   = j0 + q;
      const int gyq = gy0 + q;
      if (xedge || gyq == 0 || gyq == NYc - 1) {
        // Rare domain-edge path: exact one-sided 2nd-order stencils from global.
        float df = slow_df(fb, A0, A1, B0, B1, B2, gx, gyq);
        res[q] = fmaxf(fmaf(df, dtb, fb[gx * NYc + gyq]), 0.0f);
      } else {
        // Interior fast path: all operands from LDS; consecutive-j reads merge to b64/b128.
        float Fc  = SM(foff, i,     j);
        float Fxm = SM(foff, i - 1, j),     Fxp = SM(foff, i + 1, j);
        float Fym = SM(foff, i,     j - 1), Fyp = SM(foff, i,     j + 1);
        float Fmm = SM(foff, i - 1, j - 1), Fmp = SM(foff, i - 1, j + 1);
        float Fpm = SM(foff, i + 1, j - 1), Fpp = SM(foff, i + 1, j + 1);

        float gA = 0.5f * (SM(OFF_A0, i + 1, j) * Fxp - SM(OFF_A0, i - 1, j) * Fxm)
                 + 0.5f * (SM(OFF_A1, i, j + 1) * Fyp - SM(OFF_A1, i, j - 1) * Fym);

        float g2 = (SM(OFF_B0, i + 1, j) * Fxp - 2.f * SM(OFF_B0, i, j) * Fc
                    + SM(OFF_B0, i - 1, j) * Fxm)
                 + (SM(OFF_B1, i, j + 1) * Fyp - 2.f * SM(OFF_B1, i, j) * Fc
                    + SM(OFF_B1, i, j - 1) * Fym)
                 // 2 * dxdy: 2 * 0.25 * (Ppp - Ppm - Pmp + Pmm)
                 + 0.5f * (SM(OFF_B2, i + 1, j + 1) * Fpp - SM(OFF_B2, i + 1, j - 1) * Fpm
                         - SM(OFF_B2, i - 1, j + 1) * Fmp + SM(OFF_B2, i - 1, j - 1) * Fmm);

        float df = 0.5f * g2 - gA;
        res[q] = fmaxf(fmaf(df, dtb, Fc), 0.0f);
      }
    }
    // 16B-aligned coalesced vector store (gy0 is a multiple of 4).
    *reinterpret_cast<float4*>(&ob[gx * NYc + gy0]) =
        make_float4(res[0], res[1], res[2], res[3]);

    // Handshake: my async loads for b+1 done, then all waves' loads done.
    wait_async_0();
    __syncthreads();
  }
}

extern "C" void kernel_launch(void* const* d_in, const int* in_sizes, int n_in,
                              void* d_out, int out_size, void* d_ws, size_t ws_size,
                              hipStream_t stream) {
  (void)in_sizes; (void)n_in; (void)out_size; (void)d_ws; (void)ws_size;
  const float* f  = (const float*)d_in[0];
  const float* Ag = (const float*)d_in[1];
  const float* Bg = (const float*)d_in[2];
  const float* dt = (const float*)d_in[3];
  float* out = (float*)d_out;

  dim3 grid(NYc / TY, NXc / TX);   // (16, 64) = 1024 blocks, 8 wave32 each
  fp2d_step<<<grid, TPB, 0, stream>>>(f, Ag, Bg, dt, out);
}